// QNN_Exp_77824807403809
// MI455X (gfx1250) — compile-verified
//
#include <hip/hip_runtime.h>

typedef float v2f __attribute__((ext_vector_type(2)));
typedef float v8f __attribute__((ext_vector_type(8)));

// ---------------- small complex helpers (fp32, matches reference complex64) ----
struct C2 { float re, im; };
__device__ __forceinline__ C2 cmul(C2 a, C2 b) {
    return { a.re*b.re - a.im*b.im, a.re*b.im + a.im*b.re };
}
__device__ __forceinline__ C2 cadd(C2 a, C2 b) { return { a.re+b.re, a.im+b.im }; }

// ---------------- kernel 1: fold theta -> 6 affine constants -------------------
// Propagates basis columns e0 and e2 through the 6-layer circuit; then
// z_j(x) = K0_j + K1_j*cos(x) + K2_j*sin(x). k[j*3 + {0,1,2}] = {K0,K1,K2}_j.
__global__ void qnn_constants_kernel(const float* __restrict__ theta,
                                     float* __restrict__ k)
{
    if (threadIdx.x != 0 || blockIdx.x != 0) return;

    C2 A [4] = { {1.f,0.f},{0.f,0.f},{0.f,0.f},{0.f,0.f} };  // W * |00>
    C2 Bv[4] = { {0.f,0.f},{0.f,0.f},{1.f,0.f},{0.f,0.f} };  // W * |10>

    for (int l = 0; l < 6; ++l) {
        for (int w = 0; w < 2; ++w) {
            float phi = theta[l*6 + w*3 + 0];
            float th  = theta[l*6 + w*3 + 1];
            float omg = theta[l*6 + w*3 + 2];
            float c = cosf(0.5f*th), s = sinf(0.5f*th);
            float pp = 0.5f*(phi+omg), pm = 0.5f*(phi-omg);
            C2 ep = { cosf(pp), -sinf(pp) };   // e^{-i(phi+omega)/2}
            C2 em = { cosf(pm), -sinf(pm) };   // e^{-i(phi-omega)/2}
            C2 U00 = {  ep.re*c,  ep.im*c };
            C2 U01 = { -em.re*s,  em.im*s };   // -conj(em)*s
            C2 U10 = {  em.re*s,  em.im*s };
            C2 U11 = {  ep.re*c, -ep.im*c };   // conj(ep)*c

            C2* V[2] = { A, Bv };
            for (int v = 0; v < 2; ++v) {
                C2* st = V[v];
                if (w == 0) {                       // U on wire0: pairs (q, q+2)
                    for (int q = 0; q < 2; ++q) {
                        C2 a0 = st[q], a1 = st[q+2];
                        st[q]   = cadd(cmul(U00,a0), cmul(U01,a1));
                        st[q+2] = cadd(cmul(U10,a0), cmul(U11,a1));
                    }
                } else {                            // U on wire1: pairs (2q, 2q+1)
                    for (int q = 0; q < 2; ++q) {
                        C2 a0 = st[2*q], a1 = st[2*q+1];
                        st[2*q]   = cadd(cmul(U00,a0), cmul(U01,a1));
                        st[2*q+1] = cadd(cmul(U10,a0), cmul(U11,a1));
                    }
                }
            }
        }
        // CNOT(0,1): swap idx 2,3 ; CNOT(1,0): swap idx 1,3
        for (int v = 0; v < 2; ++v) {
            C2* st = (v == 0) ? A : Bv;
            C2 t = st[2]; st[2] = st[3]; st[3] = t;
            t = st[1]; st[1] = st[3]; st[3] = t;
        }
    }

    const float s0[4] = { 1.f,  1.f, -1.f, -1.f };   // Z on wire0
    const float s1[4] = { 1.f, -1.f,  1.f, -1.f };   // Z on wire1
    for (int j = 0; j < 2; ++j) {
        const float* sg = j ? s1 : s0;
        float al = 0.f, be = 0.f, ga = 0.f;
        for (int q = 0; q < 4; ++q) {
            al += sg[q] * (A [q].re*A [q].re + A [q].im*A [q].im);
            be += sg[q] * (Bv[q].re*Bv[q].re + Bv[q].im*Bv[q].im);
            ga -= sg[q] * (A[q].im*Bv[q].re - A[q].re*Bv[q].im);
        }
        k[j*3 + 0] = 0.5f*(al + be);   // K0
        k[j*3 + 1] = 0.5f*(al - be);   // K1 (multiplies cos x)
        k[j*3 + 2] = ga;               // K2 (multiplies sin x)
    }
}

// ---------------- kernel 2: streaming evaluation via V_WMMA_F32_16X16X4_F32 ----
// Per 16-sample tile: D(16x16) = A(16x4 consts) x B(4x16 features).
//   A rows 0,1 (and duplicated into rows 8,9) = [K0_j, K1_j, K2_j, 0]
//   B col n = [1, cos x_n, sin x_n, 0]^T
// With rows 8,9 duplicated, D lanes 16-31 (M=8,9) mirror lanes 0-15 (M=0,1),
// so EVERY lane holds (z0,z1) of sample N = lane&15. Tiles are processed in
// pairs: low half stores tile0, high half stores tile1 -> one full-width
// 32-lane global_store_b64 per 32 samples, no EXEC masking.
#define PAIRS_PER_WAVE  4
#define WAVES_PER_BLOCK 8
#define SAMPLES_PER_BLOCK (WAVES_PER_BLOCK * PAIRS_PER_WAVE * 32)   // 1024

#define INV_2PI 0.15915494309189535f

__global__ void __launch_bounds__(256)
qnn_eval_wmma_kernel(const float* __restrict__ x,
                     const float* __restrict__ k,
                     float* __restrict__ out, int n)
{
    const int lane = threadIdx.x & 31;
    const int wave = threadIdx.x >> 5;
    const int m    = lane & 15;          // A row / sample-in-tile
    const bool hi  = (lane & 16) != 0;   // K half select

    const int base = blockIdx.x * SAMPLES_PER_BLOCK
                   + wave * (PAIRS_PER_WAVE * 32);

    if (base + WAVES_PER_BLOCK * 0 >= 0 && base + SAMPLES_PER_BLOCK <= n &&
        blockIdx.x * SAMPLES_PER_BLOCK + SAMPLES_PER_BLOCK <= n) {
        // ---------------- fast path: no guards, WMMA pipeline ----------------
        // A-matrix (constants): rows 0,1 and 8,9 carry the coefficients.
        v2f a; a[0] = 0.0f; a[1] = 0.0f;
        const int r = m & 7;
        if (r < 2) {
            a[0] = hi ? k[r*3 + 2] : k[r*3 + 0];   // K=2 : K=0
            a[1] = hi ? 0.0f       : k[r*3 + 1];   // K=3 : K=1
        }

        const float bias = hi ? 0.0f : 0.25f;      // cos_rev(r) = sin_rev(r+1/4)
        const float* xp = x + base + m;            // both halves read sample g+m
        float2* op = reinterpret_cast<float2*>(out) + base + lane;

#pragma unroll
        for (int p = 0; p < PAIRS_PER_WAVE; ++p) {
            const float xv0 = xp[p*32];            // tile 2p
            const float xv1 = xp[p*32 + 16];       // tile 2p+1

            // one TRANS op per lane per tile (hi: sin, lo: cos)
            const float t0 = __builtin_amdgcn_sinf(__builtin_fmaf(xv0, INV_2PI, bias));
            const float t1 = __builtin_amdgcn_sinf(__builtin_fmaf(xv1, INV_2PI, bias));

            v2f b0, b1;
            b0[0] = hi ? t0 : 1.0f;  b0[1] = hi ? 0.0f : t0;
            b1[0] = hi ? t1 : 1.0f;  b1[1] = hi ? 0.0f : t1;

            v8f c = { 0.f, 0.f, 0.f, 0.f, 0.f, 0.f, 0.f, 0.f };
            v8f d0 = __builtin_amdgcn_wmma_f32_16x16x4_f32(
                        false, a, false, b0, (short)0, c, false, false);
            v8f d1 = __builtin_amdgcn_wmma_f32_16x16x4_f32(
                        false, a, false, b1, (short)0, c, false, false);

            // low half lanes -> tile0 results, high half lanes -> tile1 results
            float2 rzz;
            rzz.x = hi ? d1[0] : d0[0];
            rzz.y = hi ? d1[1] : d0[1];
            op[p*32] = rzz;                        // full 32-lane b64 store
        }
    } else {
        // ---------------- guarded fallback (partial block; plain VALU) -------
        const float k00 = k[0], k01 = k[1], k02 = k[2];
        const float k10 = k[3], k11 = k[4], k12 = k[5];
        const int blkBase = blockIdx.x * SAMPLES_PER_BLOCK;
        for (int i = blkBase + (int)threadIdx.x;
             i < n && i < blkBase + SAMPLES_PER_BLOCK; i += 256) {
            const float xv = x[i];
            const float cv = __builtin_amdgcn_sinf(__builtin_fmaf(xv, INV_2PI, 0.25f));
            const float sv = __builtin_amdgcn_sinf(xv * INV_2PI);
            out[2*i + 0] = k00 + k01*cv + k02*sv;
            out[2*i + 1] = k10 + k11*cv + k12*sv;
        }
    }
}

// ---------------- launcher -----------------------------------------------------
extern "C" void kernel_launch(void* const* d_in, const int* in_sizes, int n_in,
                              void* d_out, int out_size, void* d_ws, size_t ws_size,
                              hipStream_t stream) {
    const float* x     = (const float*)d_in[0];   // [B] float32
    const float* theta = (const float*)d_in[1];   // [6,2,3] float32
    float*       out   = (float*)d_out;           // [B,2] float32
    float*       kbuf  = (float*)d_ws;            // 6 floats scratch

    const int n = in_sizes[0];

    qnn_constants_kernel<<<1, 32, 0, stream>>>(theta, kbuf);

    const int blocks = (n + SAMPLES_PER_BLOCK - 1) / SAMPLES_PER_BLOCK;
    qnn_eval_wmma_kernel<<<blocks, 256, 0, stream>>>(x, kbuf, out, n);
}